// Encoder_66657892434368
// MI455X (gfx1250) — compile-verified
//
#include <hip/hip_runtime.h>

// ---------------------------------------------------------------------------
// GraphConvolution: out = segment_sum(support[edge_src] * edge_w, edge_dst) + b
//                   support = x @ W
// N=100000, E=1600000, D_IN=256, D_OUT=128, fp32.
// GEMM is HBM-bound (x: 102.4 MB single pass) -> fp32 WMMA, x staged in LDS
// via gfx1250 async global->LDS so each x byte crosses HBM exactly once.
// ---------------------------------------------------------------------------

#define D_IN  256
#define D_OUT 128
#define ROWS_PER_BLOCK 32                       // 2 M-subtiles of 16 per wave
#define TILE_FLOATS (ROWS_PER_BLOCK * D_IN)     // 8192 floats = 32 KB LDS

typedef __attribute__((ext_vector_type(2))) float v2f;
typedef __attribute__((ext_vector_type(8))) float v8f;
typedef __attribute__((ext_vector_type(4))) int  v4i;

typedef __attribute__((address_space(1))) v4i* gv4i_p;   // global (device) AS
typedef __attribute__((address_space(3))) v4i* lv4i_p;   // LDS AS

#if defined(__has_builtin)
#if __has_builtin(__builtin_amdgcn_global_load_async_to_lds_b128)
#define HAVE_ASYNC_LDS 1
#endif
#endif

__device__ __forceinline__ void wait_asynccnt0() {
#if defined(__has_builtin) && __has_builtin(__builtin_amdgcn_s_wait_asynccnt)
    __builtin_amdgcn_s_wait_asynccnt(0);
#else
    asm volatile("s_wait_asynccnt 0x0" ::: "memory");
#endif
}

// ---------------------------------------------------------------------------
// Kernel 1: support = x @ W with V_WMMA_F32_16X16X4_F32.
// Block: 256 thr = 8 waves, owns rows [32b, 32b+32), wave w owns cols [16w,16w+16).
// Per ISA 7.12.2 fp32 layouts (wave32):
//   A 16x4 : lane half h, row m=lane&15, holds K = 2h + {0,1}
//   B 4x16 : comp v, half h -> K = v + 2h, N = lane&15
//   C 16x16: comp v, half h -> M = v + 8h, N = lane&15
// ---------------------------------------------------------------------------
__global__ __launch_bounds__(256)
void gcn_gemm_wmma(const float* __restrict__ x,
                   const float* __restrict__ W,
                   float* __restrict__ support,
                   int n_nodes) {
    __shared__ float tile[TILE_FLOATS];         // 32 rows x 256 cols, row-major

    const int tid  = threadIdx.x;
    const int lane = tid & 31;
    const int wave = tid >> 5;
    const int half = lane >> 4;                 // 0 or 1
    const int m    = lane & 15;
    const int row0 = blockIdx.x * ROWS_PER_BLOCK;
    const int n0   = wave << 4;

    // ---- Stage x tile into LDS: 2048 float4 chunks, 8 per thread ----------
#if defined(HAVE_ASYNC_LDS)
    for (int i = tid; i < TILE_FLOATS / 4; i += 256) {
        int r = i >> 6;                         // 64 float4 per row
        int c = i & 63;
        int gr = row0 + r;
        if (gr >= n_nodes) gr = n_nodes - 1;    // clamp (exact tiling: no-op)
        __builtin_amdgcn_global_load_async_to_lds_b128(
            (gv4i_p)(x + (size_t)gr * D_IN + (c << 2)),
            (lv4i_p)(&tile[i << 2]),
            0, 0);
    }
    wait_asynccnt0();
    __syncthreads();
#else
    for (int i = tid; i < TILE_FLOATS / 4; i += 256) {
        int r = i >> 6;
        int c = i & 63;
        int gr = row0 + r;
        if (gr >= n_nodes) gr = n_nodes - 1;
        float4 v = *reinterpret_cast<const float4*>(x + (size_t)gr * D_IN + (c << 2));
        *reinterpret_cast<float4*>(&tile[i << 2]) = v;
    }
    __syncthreads();
#endif

    // ---- K loop: B frag loaded once, reused for 2 M-subtiles --------------
    const float* __restrict__ bcol = W + (size_t)(2 * half) * D_OUT + n0 + m;
    const float* a0p = &tile[(size_t)m * D_IN + 2 * half];          // rows 0..15
    const float* a1p = &tile[(size_t)(16 + m) * D_IN + 2 * half];   // rows 16..31

    v8f acc0 = {};
    v8f acc1 = {};

#pragma unroll 4
    for (int k = 0; k < D_IN; k += 4) {
        v2f b;
        b.x = bcol[(size_t)k * D_OUT];          // B[k + 2h][n0+m]
        b.y = bcol[(size_t)(k + 1) * D_OUT];    // B[k + 2h + 1][n0+m]
        v2f a0 = *reinterpret_cast<const v2f*>(a0p + k);   // ds_load_b64
        v2f a1 = *reinterpret_cast<const v2f*>(a1p + k);
        acc0 = __builtin_amdgcn_wmma_f32_16x16x4_f32(
            false, a0, false, b, (short)0, acc0, false, false);
        acc1 = __builtin_amdgcn_wmma_f32_16x16x4_f32(
            false, a1, false, b, (short)0, acc1, false, false);
    }

    // ---- Store C/D fragments ---------------------------------------------
    float* __restrict__ o0 = support + (size_t)(row0 + 8 * half) * D_OUT + n0 + m;
    if (row0 + 16 <= n_nodes) {                  // wave-uniform guard
#pragma unroll
        for (int v = 0; v < 8; ++v) o0[(size_t)v * D_OUT] = acc0[v];
    }
    float* __restrict__ o1 = o0 + (size_t)16 * D_OUT;
    if (row0 + 32 <= n_nodes) {
#pragma unroll
        for (int v = 0; v < 8; ++v) o1[(size_t)v * D_OUT] = acc1[v];
    }
}

// ---------------------------------------------------------------------------
// Kernel 2: out[n][j] = b[j]  (seed accumulator with bias; fuses the "+ b")
// ---------------------------------------------------------------------------
__global__ __launch_bounds__(256)
void gcn_bias_init(const float* __restrict__ bias,
                   float* __restrict__ out,
                   int total) {
    int i = blockIdx.x * 256 + threadIdx.x;
    if (i < total) out[i] = bias[i & (D_OUT - 1)];
}

// ---------------------------------------------------------------------------
// Kernel 3: one wave per edge. Lane L: float4 gather (global_load_b128) from
// support[src], scale by wave-uniform weight, 4x global_atomic_add_f32 into
// out[dst]. support (51.2 MB) + out (51.2 MB) are L2-resident (192 MB).
// ---------------------------------------------------------------------------
__global__ __launch_bounds__(256)
void gcn_edge_scatter(const float* __restrict__ support,
                      const int* __restrict__ edge_src,
                      const int* __restrict__ edge_dst,
                      const float* __restrict__ edge_w,
                      float* __restrict__ out,
                      int n_edges) {
    const int lane = threadIdx.x & 31;
    const int e = blockIdx.x * 8 + (threadIdx.x >> 5);   // 8 waves per block
    if (e >= n_edges) return;

    const int s = edge_src[e];
    const int d = edge_dst[e];
    const float w = edge_w[e];

    const float4* __restrict__ srow =
        reinterpret_cast<const float4*>(support + (size_t)s * D_OUT);
    float4 v = srow[lane];

    float* __restrict__ drow = out + (size_t)d * D_OUT + lane * 4;
    unsafeAtomicAdd(drow + 0, v.x * w);
    unsafeAtomicAdd(drow + 1, v.y * w);
    unsafeAtomicAdd(drow + 2, v.z * w);
    unsafeAtomicAdd(drow + 3, v.w * w);
}

// ---------------------------------------------------------------------------
// Launch
// ---------------------------------------------------------------------------
extern "C" void kernel_launch(void* const* d_in, const int* in_sizes, int n_in,
                              void* d_out, int out_size, void* d_ws, size_t ws_size,
                              hipStream_t stream) {
    const float* x        = (const float*)d_in[0];
    const int*   edge_src = (const int*)  d_in[1];
    const int*   edge_dst = (const int*)  d_in[2];
    const float* edge_w   = (const float*)d_in[3];
    const float* W        = (const float*)d_in[4];
    const float* bias     = (const float*)d_in[5];
    float* out = (float*)d_out;

    const int n_nodes = in_sizes[0] / D_IN;     // 100000
    const int n_edges = in_sizes[1];            // 1600000
    const int total_out = n_nodes * D_OUT;

    float* support = (float*)d_ws;              // 51.2 MB scratch

    // 1) support = x @ W  (fp32 WMMA, x staged via async global->LDS)
    const int m_blocks = (n_nodes + ROWS_PER_BLOCK - 1) / ROWS_PER_BLOCK;  // 3125
    gcn_gemm_wmma<<<m_blocks, 256, 0, stream>>>(x, W, support, n_nodes);

    // 2) out = broadcast(b)
    gcn_bias_init<<<(total_out + 255) / 256, 256, 0, stream>>>(bias, out, total_out);

    // 3) out += segment_sum(support[src] * w, dst)
    gcn_edge_scatter<<<(n_edges + 7) / 8, 256, 0, stream>>>(
        support, edge_src, edge_dst, edge_w, out, n_edges);
}